// BucketConvLayerCUDA_73057393705081
// MI455X (gfx1250) — compile-verified
//
#include <hip/hip_runtime.h>

#define HH 256
#define WW 256
#define BB 4
#define CIN 8
#define COUT 8
#define NTYPES 216
#define NPIX (BB*HH*WW)          // 262144
#define KDIM 72                   // Cin*K*K
#define NW   (COUT*KDIM)          // 576
#define EMB_STRIDE (NW + COUT)    // 584

typedef float v2f __attribute__((ext_vector_type(2)));
typedef float v8f __attribute__((ext_vector_type(8)));

__global__ void k_zero_counts(int* counts) {
    int i = blockIdx.x * blockDim.x + threadIdx.x;
    if (i < NTYPES) counts[i] = 0;
}

__global__ void k_hist(const int* __restrict__ buckets, int* __restrict__ counts) {
    int i = blockIdx.x * blockDim.x + threadIdx.x;
    if (i < NPIX) atomicAdd(&counts[buckets[i]], 1);
}

__global__ void k_scan(const int* __restrict__ counts, int* __restrict__ offsets,
                       int* __restrict__ cursor) {
    int acc = 0;
    for (int t = 0; t < NTYPES; ++t) {
        offsets[t] = acc;
        cursor[t]  = acc;
        acc += counts[t];
    }
    offsets[NTYPES] = acc;
}

__global__ void k_scatter(const int* __restrict__ buckets, int* __restrict__ cursor,
                          int* __restrict__ perm) {
    int i = blockIdx.x * blockDim.x + threadIdx.x;
    if (i < NPIX) {
        int p = atomicAdd(&cursor[buckets[i]], 1);
        perm[p] = i;
    }
}

#define WAVES_PER_BLOCK 8
#define BLOCKS_PER_TYPE 4

// One bucket per blockIdx.x; each wave processes 16-pixel tiles of that bucket.
// Per tile: D(16x16) = sum_{cc=0..17} A(16x4) * B(4x16) via v_wmma_f32_16x16x4_f32.
// A rows = im2col patches of 16 same-bucket pixels; B cols 0..7 = bucket weights, 8..15 = 0.
__global__ __launch_bounds__(256) void k_bucket_gemm(
    const float* __restrict__ x, const float* __restrict__ emb,
    const int* __restrict__ offsets, const int* __restrict__ perm,
    float* __restrict__ out) {

    const int t    = blockIdx.x;
    const int lane = threadIdx.x & 31;
    const int gwave  = blockIdx.y * WAVES_PER_BLOCK + (threadIdx.x >> 5);
    const int nwaves = gridDim.y * WAVES_PER_BLOCK;

    const int start  = offsets[t];
    const int end    = offsets[t + 1];
    const int ntiles = (end - start + 15) >> 4;

    const float* wrow = emb + t * EMB_STRIDE;

    // ---- B matrix (shared across all tiles of this bucket): 18 chunks of 4x16 ----
    // f32 layout: VGPR0: lanes 0-15 -> K=k0+0,N=lane; lanes 16-31 -> K=k0+2,N=lane-16
    //             VGPR1: K=k0+1 / K=k0+3
    const int n  = lane & 15;          // output column
    const int hi = lane >> 4;          // lane half (K offset +2)
    const int khalf = hi << 1;
    v2f bmat[18];
    #pragma unroll
    for (int cc = 0; cc < 18; ++cc) {
        const int k = cc * 4 + khalf;
        float b0 = 0.f, b1 = 0.f;
        if (n < COUT) {                // emb row stores w[o][k] at o*72 + k
            b0 = wrow[n * KDIM + k];
            b1 = wrow[n * KDIM + k + 1];
        }
        bmat[cc].x = b0; bmat[cc].y = b1;
    }
    const float biasv = (n < COUT) ? wrow[NW + n] : 0.f;

    for (int tile = gwave; tile < ntiles; tile += nwaves) {
        const int rowbase = start + tile * 16;

        // Row pixel for this lane (both halves hold rows 0..15 of the tile).
        const int m     = lane & 15;
        const int pidx  = rowbase + m;
        const int valid = (pidx < end) ? 1 : 0;
        const int cp    = valid ? pidx : start;   // clamped, always legal
        const int pix   = perm[cp];               // b*65536 + h*256 + w
        const int pb = pix >> 16;
        const int ph = (pix >> 8) & 255;
        const int pw = pix & 255;

        // ---- 9 distinct tap offsets/masks per pixel (shared by all 72 k's) ----
        const float rowmask = valid ? 1.f : 0.f;
        int   hrow[3], wcol[3];
        float hmask[3], wmask[3];
        #pragma unroll
        for (int d = 0; d < 3; ++d) {
            const int hh = ph + d - 1;
            const int ww = pw + d - 1;
            hrow[d]  = (hh & 255) << 8;            // wrapped: address always in-bounds
            wcol[d]  = ww & 255;
            hmask[d] = ((unsigned)hh < 256u) ? 1.f : 0.f;
            wmask[d] = ((unsigned)ww < 256u) ? 1.f : 0.f;
        }
        const int base = (pb * CIN) << 16;
        int   off9[9];
        float msk9[9];
        #pragma unroll
        for (int r = 0; r < 9; ++r) {
            off9[r] = base + hrow[r / 3] + wcol[r % 3];
            msk9[r] = hmask[r / 3] * wmask[r % 3] * rowmask;
        }

        // ---- Per-lane rotated tap table: half-B tap = half-A tap rotated by +2 mod 9 ----
        // (kB % 9 == (kA + 2) % 9), so one 18-select rotation replaces 72 per-element selects.
        int   offR[9];
        float mskR[9];
        #pragma unroll
        for (int s = 0; s < 9; ++s) {
            const int s2 = (s + 2) % 9;
            offR[s] = hi ? off9[s2] : off9[s];
            mskR[s] = hi ? msk9[s2] : msk9[s];
        }

        // ---- Gather all A chunks: constant-indexed table refs, load + masked mul ----
        v2f a[18];
        #pragma unroll
        for (int cc = 0; cc < 18; ++cc) {
            #pragma unroll
            for (int j = 0; j < 2; ++j) {
                const int kbase = cc * 4 + j;          // this lane's k is kbase + khalf
                const int tapc  = kbase % 9;           // rotated table absorbs khalf
                const int cA = (kbase / 9) << 16;
                const int cB = ((kbase + 2) / 9) << 16;
                // channel offset: differs between halves only on tap carry (tapc >= 7)
                const int coff = (cA == cB) ? cA : (hi ? cB : cA);
                const float v = x[offR[tapc] + coff] * mskR[tapc];
                if (j == 0) a[cc].x = v; else a[cc].y = v;
            }
        }

        // ---- WMMA chain: 18 back-to-back fp32 outer-product accumulates ----
        v8f acc = {};
        #pragma unroll
        for (int cc = 0; cc < 18; ++cc) {
            acc = __builtin_amdgcn_wmma_f32_16x16x4_f32(
                false, a[cc], false, bmat[cc], (short)0, acc, false, false);
        }

        // ---- D: VGPR r holds rows r (lanes 0-15) and r+8 (lanes 16-31), N = lane&15 ----
        const int col  = lane & 15;
        const int rofs = hi << 3;
        #pragma unroll
        for (int r = 0; r < 8; ++r) {
            const int row   = r + rofs;
            const int pix2  = __shfl(pix, row, 32);   // lane `row` holds that row's pixel
            if (col < COUT && (rowbase + row) < end) {
                const int b2 = pix2 >> 16;
                const int h2 = (pix2 >> 8) & 255;
                const int w2 = pix2 & 255;
                out[((b2 * COUT + col) << 16) + (h2 << 8) + w2] = acc[r] + biasv;
            }
        }
    }
}

extern "C" void kernel_launch(void* const* d_in, const int* in_sizes, int n_in,
                              void* d_out, int out_size, void* d_ws, size_t ws_size,
                              hipStream_t stream) {
    const float* x       = (const float*)d_in[0];
    const float* emb     = (const float*)d_in[1];
    const int*   buckets = (const int*)d_in[2];
    float*       out     = (float*)d_out;

    // Workspace layout
    int* counts  = (int*)d_ws;                          // 216 ints
    int* cursor  = (int*)((char*)d_ws + 1024);          // 216 ints
    int* offsets = (int*)((char*)d_ws + 2048);          // 217 ints
    int* perm    = (int*)((char*)d_ws + 4096);          // NPIX ints

    k_zero_counts<<<1, 256, 0, stream>>>(counts);
    k_hist<<<(NPIX + 255) / 256, 256, 0, stream>>>(buckets, counts);
    k_scan<<<1, 1, 0, stream>>>(counts, offsets, cursor);
    k_scatter<<<(NPIX + 255) / 256, 256, 0, stream>>>(buckets, cursor, perm);
    k_bucket_gemm<<<dim3(NTYPES, BLOCKS_PER_TYPE), 256, 0, stream>>>(
        x, emb, offsets, perm, out);
}